// SoftBernoulliGraphVariationalAutoencoder_66597762892109
// MI455X (gfx1250) — compile-verified
//
#include <hip/hip_runtime.h>
#include <hip/hip_bf16.h>

typedef float v2f __attribute__((ext_vector_type(2)));
typedef float v8f __attribute__((ext_vector_type(8)));

#define D 128      // D_IN == F == 128
#define SLABS 4    // 16-row slabs per block in the GEMM (B kept in registers)

// ---------------- utility kernels ----------------
__global__ void zero_kernel(float* __restrict__ p, long long count) {
  long long i = (long long)blockIdx.x * blockDim.x + threadIdx.x;
  long long stride = (long long)gridDim.x * blockDim.x;
  for (; i < count; i += stride) p[i] = 0.0f;
}

__global__ void deg_kernel(const int* __restrict__ src, float* __restrict__ deg, int E) {
  int e = blockIdx.x * blockDim.x + threadIdx.x;
  if (e < E) atomicAdd(&deg[src[e]], 1.0f);
}

__global__ void rsqrt_kernel(float* __restrict__ d, int N) {
  int n = blockIdx.x * blockDim.x + threadIdx.x;
  if (n < N) {
    float v = d[n];
    d[n] = (v > 0.0f) ? rsqrtf(v) : 0.0f;
  }
}

// ---------------- dense projection: Out = (A * norm^npow) @ W  ----------------
// Block = 256 threads = 8 waves. Each block computes SLABS*16 rows x all 128
// cols. Wave w owns columns [16w, 16w+16). Its B fragments (W[:, 16w:16w+16])
// are loaded ONCE into 64 VGPRs and reused across all SLABS row-slabs; the
// A slab streams through LDS. 32 x V_WMMA_F32_16X16X4_F32 per slab per wave.
__global__ __launch_bounds__(256)
void gemm_norm_wmma(const float* __restrict__ A, const float* __restrict__ nrm,
                    const float* __restrict__ W, float* __restrict__ Out,
                    int N, int npow) {
  __shared__ float lds_a[16][D];

  const int t    = threadIdx.x;
  const int wave = t >> 5;     // 0..7 -> column tile
  const int lane = t & 31;
  const int lm   = lane & 15;  // position within 16-wide group
  const int half = lane >> 4;  // 0 or 1
  const int n0   = wave * 16;

  // Register-resident B fragments for this wave's 16 columns.
  // Fragment i covers K = [4i, 4i+4): VGPR0 = K (k), VGPR1 = K+1, where
  // lanes 0-15 take K=4i..4i+1 and lanes 16-31 take K=4i+2..4i+3 (mirror of
  // the documented A-fragment layout), N = n0 + lm.
  v2f bfrag[D / 4];
#pragma unroll
  for (int i = 0; i < D / 4; ++i) {
    const int k = 4 * i + 2 * half;
    bfrag[i].x = W[(size_t)k * D + n0 + lm];
    bfrag[i].y = W[(size_t)(k + 1) * D + n0 + lm];
  }

  const int slab_base = blockIdx.x * (16 * SLABS);

  for (int s = 0; s < SLABS; ++s) {
    const int m0 = slab_base + s * 16;

    __syncthreads();  // previous slab's LDS reads complete before overwrite
    {
      // Cooperative load of the 16 x 128 A-slab, scaled by norm (or norm^2).
      int r  = t >> 4;         // 0..15 row within slab
      int c0 = (t & 15) * 8;   // 8 floats per thread
      int row = m0 + r;
      if (row < N) {
        float sc = nrm[row];
        if (npow == 2) sc *= sc;
        const float4 x = *(const float4*)(A + (size_t)row * D + c0);
        const float4 y = *(const float4*)(A + (size_t)row * D + c0 + 4);
        lds_a[r][c0 + 0] = x.x * sc;  lds_a[r][c0 + 1] = x.y * sc;
        lds_a[r][c0 + 2] = x.z * sc;  lds_a[r][c0 + 3] = x.w * sc;
        lds_a[r][c0 + 4] = y.x * sc;  lds_a[r][c0 + 5] = y.y * sc;
        lds_a[r][c0 + 6] = y.z * sc;  lds_a[r][c0 + 7] = y.w * sc;
      } else {
        for (int j = 0; j < 8; ++j) lds_a[r][c0 + j] = 0.0f;
      }
    }
    __syncthreads();

    v8f acc = {};
#pragma unroll
    for (int i = 0; i < D / 4; ++i) {
      const int k = 4 * i + 2 * half;
      // A 16x4 fragment per ISA layout: VGPR0 = K, VGPR1 = K+1; lanes 0-15
      // hold K=4i..4i+1, lanes 16-31 hold K=4i+2..4i+3; M = lm.
      v2f a;
      a.x = lds_a[lm][k];
      a.y = lds_a[lm][k + 1];
      acc = __builtin_amdgcn_wmma_f32_16x16x4_f32(
          /*neg_a=*/false, a, /*neg_b=*/false, bfrag[i],
          /*c_mod=*/(short)0, acc, /*reuse_a=*/false, /*reuse_b=*/false);
    }

    // C/D layout: VGPR v -> row m0 + v + 8*half, col n0 + lm.
#pragma unroll
    for (int v = 0; v < 8; ++v) {
      int row = m0 + v + 8 * half;
      if (row < N) Out[(size_t)row * D + n0 + lm] = acc[v];
    }
  }
}

// ---------------- edge scatter: Acc[src] += P[dst]  (one wave per edge) -------
__global__ __launch_bounds__(256)
void scatter_kernel(const float* __restrict__ P, const int* __restrict__ src,
                    const int* __restrict__ dst, float* __restrict__ Acc, int E) {
  int w = blockIdx.x * (blockDim.x >> 5) + (threadIdx.x >> 5);
  int lane = threadIdx.x & 31;
  if (w >= E) return;
  int s = src[w];
  int d = dst[w];
  const float4 v = *(const float4*)(P + (size_t)d * D + lane * 4);
  float* base = Acc + (size_t)s * D + lane * 4;
  atomicAdd(base + 0, v.x);
  atomicAdd(base + 1, v.y);
  atomicAdd(base + 2, v.z);
  atomicAdd(base + 3, v.w);
}

// ---------------- head weight: row-sum of w_rate0 (128 threads) -----------
__global__ void rowsum_kernel(const float* __restrict__ Wr0, float* __restrict__ vr0) {
  int k = blockIdx.x * blockDim.x + threadIdx.x;
  if (k < D) {
    float s = 0.0f;
    for (int f = 0; f < D; ++f) s += Wr0[(size_t)k * D + f];
    vr0[k] = s;
  }
}

// ---------------- per-node heads (3 matvecs fused, wave per node) -------------
__global__ __launch_bounds__(256)
void head_kernel(const float* __restrict__ S, const float* __restrict__ nrm,
                 const float* __restrict__ vr0, const float* __restrict__ wr1,
                 const float* __restrict__ wa, float* __restrict__ r0s,
                 float* __restrict__ r1, float* __restrict__ al, int N) {
  int node = blockIdx.x * (blockDim.x >> 5) + (threadIdx.x >> 5);
  int lane = threadIdx.x & 31;
  if (node >= N) return;
  float s = nrm[node];  // final post-aggregation norm fused here
  const float4 h4 = *(const float4*)(S + (size_t)node * D + lane * 4);
  const float4 a4 = *(const float4*)(vr0 + lane * 4);
  const float4 b4 = *(const float4*)(wr1 + lane * 4);
  const float4 c4 = *(const float4*)(wa  + lane * 4);
  float hx = h4.x * s, hy = h4.y * s, hz = h4.z * s, hw = h4.w * s;
  float p0 = hx * a4.x + hy * a4.y + hz * a4.z + hw * a4.w;
  float p1 = hx * b4.x + hy * b4.y + hz * b4.z + hw * b4.w;
  float p2 = hx * c4.x + hy * c4.y + hz * c4.z + hw * c4.w;
#pragma unroll
  for (int off = 16; off > 0; off >>= 1) {
    p0 += __shfl_xor(p0, off, 32);
    p1 += __shfl_xor(p1, off, 32);
    p2 += __shfl_xor(p2, off, 32);
  }
  if (lane == 0) {
    r0s[node] = p0;
    r1[node]  = p1;
    al[node]  = p2;
  }
}

// ---------------- per-edge outputs (6 x E) -----------------------------------
__global__ void edge_out_kernel(const float* __restrict__ r0s, const float* __restrict__ r1,
                                const float* __restrict__ al, const int* __restrict__ src,
                                const int* __restrict__ dst, const int* __restrict__ sfake,
                                const int* __restrict__ dfake, float* __restrict__ out,
                                int E, int N, float Ff) {
  int e = blockIdx.x * blockDim.x + threadIdx.x;
  if (e >= E) return;
  int s = src[e];
  int d = dst[e];
  // jnp indexing clamps out-of-range fake indices into [0, N-1]
  int sf = sfake[e]; sf = sf < 0 ? 0 : (sf > N - 1 ? N - 1 : sf);
  int df = dfake[e]; df = df < 0 ? 0 : (df > N - 1 ? N - 1 : df);

  float rs = r0s[s], rd = r0s[d];
  float r1s = r1[s], r1d = r1[d];
  float as = al[s],  ad = al[d];
  float r0sf = r0s[sf], r1sf = r1[sf], r1df = r1[df], adf = al[df];

  size_t ee = (size_t)e;
  out[0 * (size_t)E + ee] = expf(rs + rd);                         // rate0_real
  out[1 * (size_t)E + ee] = expf(r1s + r1d);                       // rate1_real
  out[2 * (size_t)E + ee] = 1.0f / (1.0f + expf(-(as * ad)));      // alpha_real (F=1 head)
  out[3 * (size_t)E + ee] = expf(r0sf + Ff * r1df);                // rate0_fake (broadcast sum)
  out[4 * (size_t)E + ee] = expf(r1sf + r1df);                     // rate1_fake
  out[5 * (size_t)E + ee] = 1.0f / (1.0f + expf(-(as * adf)));     // alpha_fake
}

// ---------------- driver ------------------------------------------------------
extern "C" void kernel_launch(void* const* d_in, const int* in_sizes, int n_in,
                              void* d_out, int out_size, void* d_ws, size_t ws_size,
                              hipStream_t stream) {
  const float* h    = (const float*)d_in[0];
  const float* w0   = (const float*)d_in[1];
  const float* w1   = (const float*)d_in[2];
  const float* wr0  = (const float*)d_in[3];
  const float* wr1  = (const float*)d_in[4];
  const float* wa   = (const float*)d_in[5];
  const int* esrc   = (const int*)d_in[6];
  const int* edst   = (const int*)d_in[7];
  const int* sfake  = (const int*)d_in[8];
  const int* dfake  = (const int*)d_in[9];

  const int F = in_sizes[4];        // 128 (w_rate1 has F elements)
  const int N = in_sizes[0] / F;    // D_IN == F
  const int E = in_sizes[6];
  float* out = (float*)d_out;

  // Workspace layout
  float* ws   = (float*)d_ws;
  float* nrm  = ws;                         // N (deg, then rsqrt in place)
  float* buf1 = nrm + N;                    // N*F
  float* buf2 = buf1 + (size_t)N * F;       // N*F
  float* r0s  = buf2 + (size_t)N * F;       // N
  float* r1v  = r0s + N;                    // N
  float* alv  = r1v + N;                    // N
  float* vr0  = alv + N;                    // F

  const int T = 256;
  const int gb = (N + 16 * SLABS - 1) / (16 * SLABS);

  // 1) degree -> norm = deg^-0.5
  hipLaunchKernelGGL(zero_kernel, dim3((N + T - 1) / T), dim3(T), 0, stream, nrm, (long long)N);
  hipLaunchKernelGGL(deg_kernel, dim3((E + T - 1) / T), dim3(T), 0, stream, esrc, nrm, E);
  hipLaunchKernelGGL(rsqrt_kernel, dim3((N + T - 1) / T), dim3(T), 0, stream, nrm, N);

  // 2) layer 0: P0 = (h * norm) @ W0 ; S0 = A @ P0
  hipLaunchKernelGGL(gemm_norm_wmma, dim3(gb), dim3(T), 0, stream, h, nrm, w0, buf1, N, 1);
  hipLaunchKernelGGL(zero_kernel, dim3(4096), dim3(T), 0, stream, buf2, (long long)N * F);
  hipLaunchKernelGGL(scatter_kernel, dim3((E + 7) / 8), dim3(T), 0, stream, buf1, esrc, edst, buf2, E);

  // 3) layer 1: P1 = (S0 * norm^2) @ W1 ; S1 = A @ P1   (post+pre norms fused)
  hipLaunchKernelGGL(gemm_norm_wmma, dim3(gb), dim3(T), 0, stream, buf2, nrm, w1, buf1, N, 2);
  hipLaunchKernelGGL(zero_kernel, dim3(4096), dim3(T), 0, stream, buf2, (long long)N * F);
  hipLaunchKernelGGL(scatter_kernel, dim3((E + 7) / 8), dim3(T), 0, stream, buf1, esrc, edst, buf2, E);

  // 4) heads: r0sum / rate1 / alpha matvecs with final *norm fused
  hipLaunchKernelGGL(rowsum_kernel, dim3(1), dim3(F), 0, stream, wr0, vr0);
  hipLaunchKernelGGL(head_kernel, dim3((N + 7) / 8), dim3(T), 0, stream,
                     buf2, nrm, vr0, wr1, wa, r0s, r1v, alv, N);

  // 5) per-edge outputs [6, E]
  hipLaunchKernelGGL(edge_out_kernel, dim3((E + T - 1) / T), dim3(T), 0, stream,
                     r0s, r1v, alv, esrc, edst, sfake, dfake, out, E, N, (float)F);
}